// FunctionalDiscriminator_65386582114541
// MI455X (gfx1250) — compile-verified
//
#include <hip/hip_runtime.h>
#include <hip/hip_bf16.h>
#include <stdint.h>

// Problem constants (from reference)
#define INPUT_DIM   1792
#define OUT_DIM     64
#define NBITS       14
#define NNODES      128        // INPUT_DIM / NBITS
#define TABLE_ROWS  16384      // 1 << NBITS
#define NWAVES      4          // 128 threads / wave32
#define NODES_PER_WAVE (NNODES / NWAVES)   // 32
#define CHUNK       8          // rows per async double-buffer chunk

// gfx1250 async path: s_wait_asynccnt builtin is probe-confirmed present on
// the device pass (round-1 guard passed); host pass sees 0 and takes fallback.
#if defined(__HIP_DEVICE_COMPILE__) && \
    __has_builtin(__builtin_amdgcn_s_wait_asynccnt)
#define WNN_ASYNC 1
#else
#define WNN_ASYNC 0
#endif

#if WNN_ASYNC
// Issue one async global->LDS copy of 8 bytes per lane (a full 256-B table
// row per wave-instruction). VDST operand is the 32-bit LDS byte offset: the
// low 32 bits of a generic LDS pointer are exactly that offset.
__device__ __forceinline__ void async_row_b64(const float* gsrc, float* lds_dst) {
    unsigned lds_off = (unsigned)(uintptr_t)lds_dst;
    asm volatile("global_load_async_to_lds_b64 %0, %1, off"
                 :
                 : "v"(lds_off), "v"(gsrc)
                 : "memory");
}
#endif

__global__ __launch_bounds__(128)
void wnn_discriminator_kernel(const int* __restrict__ x,
                              const int* __restrict__ mapping,
                              const float* __restrict__ table,
                              float* __restrict__ out)
{
    __shared__ int   x_sh[INPUT_DIM];
    __shared__ int   keys_sh[NNODES];
    __shared__ float partial[NWAVES][OUT_DIM];
#if WNN_ASYNC
    __shared__ float stage[NWAVES][2][CHUNK][OUT_DIM];   // 16 KB staging
#endif

    const int b    = blockIdx.x;
    const int tid  = threadIdx.x;
    const int wave = tid >> 5;
    const int lane = tid & 31;

    // ---------- Phase A: load bit row, compute 14-bit keys ----------
    const int* xrow = x + (size_t)b * INPUT_DIM;
    #pragma unroll
    for (int i = 0; i < INPUT_DIM / 128; ++i)
        x_sh[i * 128 + tid] = xrow[i * 128 + tid];
    __syncthreads();

    {
        const int* m = mapping + tid * NBITS;   // one node per thread
        int key = 0;
        #pragma unroll
        for (int j = 0; j < NBITS; ++j)
            key |= (x_sh[m[j]] & 1) << j;       // LSB-first packing
        keys_sh[tid] = key;
    }
    __syncthreads();

    // ---------- Phase B: gather table rows, accumulate ----------
    const int n0 = wave * NODES_PER_WAVE;
    float accx = 0.0f, accy = 0.0f;

#if WNN_ASYNC
    // Double-buffered async gather through LDS (ASYNCcnt-pipelined).
    #pragma unroll
    for (int r = 0; r < CHUNK; ++r) {               // prime chunk 0
        const int n = n0 + r;
        const float* row = table + ((size_t)n * TABLE_ROWS + keys_sh[n]) * OUT_DIM + lane * 2;
        async_row_b64(row, &stage[wave][0][r][lane * 2]);
    }
    #pragma unroll
    for (int c = 0; c < NODES_PER_WAVE / CHUNK; ++c) {
        const int buf = c & 1;
        if (c + 1 < NODES_PER_WAVE / CHUNK) {
            const int nb = (c + 1) & 1;
            #pragma unroll
            for (int r = 0; r < CHUNK; ++r) {
                const int n = n0 + (c + 1) * CHUNK + r;
                const float* row = table + ((size_t)n * TABLE_ROWS + keys_sh[n]) * OUT_DIM + lane * 2;
                async_row_b64(row, &stage[wave][nb][r][lane * 2]);
            }
            __builtin_amdgcn_s_wait_asynccnt(CHUNK);  // chunk c landed in LDS
        } else {
            __builtin_amdgcn_s_wait_asynccnt(0);
        }
        #pragma unroll
        for (int r = 0; r < CHUNK; ++r) {
            float2 v = *(const float2*)&stage[wave][buf][r][lane * 2];
            accx += v.x; accy += v.y;
        }
    }
#else
    // Fallback: prefetch all 32 random rows (global_prefetch_b8), then
    // coalesced float2 loads (32 lanes x 8B = one 256B row per instruction).
    #pragma unroll
    for (int i = 0; i < NODES_PER_WAVE; ++i) {
        const int n = n0 + i;
        const float* row = table + ((size_t)n * TABLE_ROWS + keys_sh[n]) * OUT_DIM;
        __builtin_prefetch(row + lane * 2, 0, 0);
    }
    #pragma unroll 8
    for (int i = 0; i < NODES_PER_WAVE; ++i) {
        const int n = n0 + i;
        const float* row = table + ((size_t)n * TABLE_ROWS + keys_sh[n]) * OUT_DIM;
        float2 v = *(const float2*)(row + lane * 2);
        accx += v.x; accy += v.y;
    }
#endif

    partial[wave][lane * 2 + 0] = accx;
    partial[wave][lane * 2 + 1] = accy;
    __syncthreads();

    // ---------- Phase C: cross-wave reduce, scale by 1/14, store ----------
    if (tid < OUT_DIM) {
        float s = partial[0][tid] + partial[1][tid] + partial[2][tid] + partial[3][tid];
        out[(size_t)b * OUT_DIM + tid] = s * (1.0f / (float)NBITS);
    }
}

extern "C" void kernel_launch(void* const* d_in, const int* in_sizes, int n_in,
                              void* d_out, int out_size, void* d_ws, size_t ws_size,
                              hipStream_t stream) {
    const int*   x       = (const int*)d_in[0];     // [4096, 1792] int32 bits
    const int*   mapping = (const int*)d_in[1];     // [1792] permutation (int32 on device)
    const float* table   = (const float*)d_in[2];   // [128, 16384, 64] f32
    float*       out     = (float*)d_out;           // [4096, 64] f32

    const int batch = out_size / OUT_DIM;           // 4096
    wnn_discriminator_kernel<<<dim3(batch), dim3(128), 0, stream>>>(x, mapping, table, out);
}